// Prototype_12884901888249
// MI455X (gfx1250) — compile-verified
//
#include <hip/hip_runtime.h>

// Problem constants (from reference): B=2, N=1024, D=30, H=64, O=30
#define ROWS 2048      // B*N
#define DD   30
#define HH   64
#define OO   30

typedef __attribute__((ext_vector_type(2))) float v2f;
typedef __attribute__((ext_vector_type(8))) float v8f;
typedef __attribute__((ext_vector_type(4))) int   v4i;

// pointer-to-AS-qualified-v4i types for the async-to-LDS builtin
typedef __attribute__((address_space(1))) v4i* gv4i_ptr;   // global
typedef __attribute__((address_space(3))) v4i* lv4i_ptr;   // LDS

#if defined(__AMDGCN__) && \
    __has_builtin(__builtin_amdgcn_global_load_async_to_lds_b128) && \
    __has_builtin(__builtin_amdgcn_s_wait_asynccnt)
#define USE_ASYNC 1
#else
#define USE_ASYNC 0
#endif

// ---------------------------------------------------------------------------
// Stage one 64x64 f32 chunk (16 KB = 1024 x b128) into LDS; 4 issues/thread.
// Async path: GLOBAL_LOAD_ASYNC_TO_LDS_B128, tracked by ASYNCcnt.
// ---------------------------------------------------------------------------
__device__ __forceinline__
void stage_chunk(const float* __restrict__ gsrc, float* ldst, int tid) {
#if USE_ASYNC
  #pragma unroll
  for (int t = 0; t < 4; ++t) {
    __builtin_amdgcn_global_load_async_to_lds_b128(
        (gv4i_ptr)(gsrc + (tid + t * 256) * 4),
        (lv4i_ptr)(ldst + (tid + t * 256) * 4),
        /*offset=*/0, /*cpol=*/0);
  }
#else
  const float4* src = (const float4*)gsrc;
  float4* dst = (float4*)ldst;
  #pragma unroll
  for (int t = 0; t < 4; ++t)
    dst[tid + t * 256] = src[tid + t * 256];
#endif
}

__device__ __forceinline__ void wait_stage(int outstanding) {
#if USE_ASYNC
  if (outstanding) __builtin_amdgcn_s_wait_asynccnt(4);
  else             __builtin_amdgcn_s_wait_asynccnt(0);
#else
  (void)outstanding;
#endif
}

// ---------------------------------------------------------------------------
// Kernel 1: c = z @ W_r[:D] + b_r   (which==0)
//           s = z @ W_r[D:]        (which==1)
// One wave per 16x16 tile. 7 unconditional K-steps + 1 masked tail step.
// Tail masking on B only (rows 30,31 -> 0); A uses clamped addresses.
// ---------------------------------------------------------------------------
__global__ void __launch_bounds__(32)
k_proj(const float* __restrict__ z,   // [ROWS, DD]
       const float* __restrict__ Wr,  // [2*DD, HH]
       const float* __restrict__ br,  // [HH]
       float* __restrict__ c,         // [ROWS, HH]
       float* __restrict__ s)         // [ROWS, HH]
{
  const int lane  = threadIdx.x;
  const int lr    = lane & 15;
  const int hi    = lane >> 4;            // 0: K=k+0,k+1 ; 1: K=k+2,k+3
  const int m0    = blockIdx.x * 16;
  const int h0    = blockIdx.y * 16;
  const int which = blockIdx.z;

  const int row = m0 + lr;
  const int col = h0 + lr;
  const float* Wbase = Wr + which * DD * HH;
  const float* zr = z + row * DD;

  v8f acc = {};
  #pragma unroll
  for (int k = 0; k < 28; k += 4) {
    const int ka = k + 2 * hi;
    v2f a; a.x = zr[ka];                a.y = zr[ka + 1];
    v2f b; b.x = Wbase[ka * HH + col];  b.y = Wbase[(ka + 1) * HH + col];
    acc = __builtin_amdgcn_wmma_f32_16x16x4_f32(
        false, a, false, b, (short)0, acc, false, false);
  }
  {
    // Tail K=28..31. hi-half lanes carry K=30,31 -> zero their B rows.
    const int kc = hi ? 0 : 28;          // clamped, in-bounds for all lanes
    const float bmask = hi ? 0.0f : 1.0f;
    v2f a; a.x = zr[kc];                a.y = zr[kc + 1];
    const float b0 = Wbase[kc * HH + col];
    const float b1 = Wbase[(kc + 1) * HH + col];
    v2f b; b.x = b0 * bmask;            b.y = b1 * bmask;
    acc = __builtin_amdgcn_wmma_f32_16x16x4_f32(
        false, a, false, b, (short)0, acc, false, false);
  }

  const float bias = (which == 0) ? br[col] : 0.0f;
  float* dst = (which == 0) ? c : s;
  #pragma unroll
  for (int v = 0; v < 8; ++v) {
    const int ro = m0 + v + 8 * hi;
    dst[ro * HH + col] = acc[v] + bias;
  }
}

// ---------------------------------------------------------------------------
// Kernel 2 (dominant): rsum[i,h] = sum_j relu(c[i,h] + s[j,h])
// 256 threads (8 wave32) per block, 16 i-rows/block. Double-buffered LDS
// staging of 64x64 s-tiles via async-to-LDS; ASYNCcnt in-order so
// wait<=4 after issuing the next chunk guarantees the current chunk landed.
// ---------------------------------------------------------------------------
__global__ void __launch_bounds__(256)
k_pair(const float* __restrict__ c,     // [ROWS, HH]
       const float* __restrict__ s,     // [ROWS, HH]
       float* __restrict__ rsum)        // [ROWS, HH]
{
  __shared__ float sj[2][64 * HH];      // 2 x 16 KB

  const int tid   = threadIdx.x;
  const int h     = tid & 63;
  const int ig    = tid >> 6;           // 0..3 -> group of 4 i-rows
  const int i0    = blockIdx.x * 16;    // tiles never straddle a batch
  const int batch = i0 >> 10;

  float cc[4], acc[4] = {0.f, 0.f, 0.f, 0.f};
  #pragma unroll
  for (int r = 0; r < 4; ++r)
    cc[r] = c[(i0 + ig * 4 + r) * HH + h];

  const float* sb = s + batch * 1024 * HH;

  stage_chunk(sb, &sj[0][0], tid);      // prefetch chunk 0

  #pragma unroll 1
  for (int ci = 0; ci < 16; ++ci) {
    const int cur = ci & 1;
    const int more = (ci + 1 < 16);
    if (more)
      stage_chunk(sb + (ci + 1) * 64 * HH, &sj[cur ^ 1][0], tid);
    wait_stage(more);                   // chunk ci fully in LDS (this wave)
    __syncthreads();                    // ... and from all waves

    const float* buf = &sj[cur][0];
    #pragma unroll 4
    for (int j = 0; j < 64; ++j) {
      const float v = buf[j * HH + h];  // conflict-free banks
      #pragma unroll
      for (int r = 0; r < 4; ++r)
        acc[r] += __builtin_fmaxf(cc[r] + v, 0.0f);
    }
    __syncthreads();                    // safe to overwrite buf next iter
  }

  #pragma unroll
  for (int r = 0; r < 4; ++r)
    rsum[(i0 + ig * 4 + r) * HH + h] = acc[r];
}

// ---------------------------------------------------------------------------
// Kernel 3: out = relu(rsum @ W_a + b_a), [2048x64]@[64x30], f32 WMMA, K=64.
// No masking in the K loop: OOB output columns are computed on a clamped
// Wa column and simply never stored.
// ---------------------------------------------------------------------------
__global__ void __launch_bounds__(32)
k_out(const float* __restrict__ rsum,  // [ROWS, HH]
      const float* __restrict__ Wa,    // [HH, OO]
      const float* __restrict__ ba,    // [OO]
      float* __restrict__ out)         // [ROWS, OO]
{
  const int lane = threadIdx.x;
  const int lr   = lane & 15;
  const int hi   = lane >> 4;
  const int m0   = blockIdx.x * 16;
  const int o0   = blockIdx.y * 16;

  const int row  = m0 + lr;
  const int col  = o0 + lr;
  const int colc = (col < OO) ? col : (OO - 1);   // clamped address
  const float* rr = rsum + row * HH;

  v8f acc = {};
  #pragma unroll
  for (int k = 0; k < HH; k += 4) {
    const int ka = k + 2 * hi;
    v2f a; a.x = rr[ka];                 a.y = rr[ka + 1];
    v2f b; b.x = Wa[ka * OO + colc];     b.y = Wa[(ka + 1) * OO + colc];
    acc = __builtin_amdgcn_wmma_f32_16x16x4_f32(
        false, a, false, b, (short)0, acc, false, false);
  }

  if (col < OO) {
    const float bias = ba[col];
    #pragma unroll
    for (int v = 0; v < 8; ++v) {
      const int ro = m0 + v + 8 * hi;
      out[ro * OO + col] = __builtin_fmaxf(acc[v] + bias, 0.0f);
    }
  }
}

// ---------------------------------------------------------------------------
extern "C" void kernel_launch(void* const* d_in, const int* in_sizes, int n_in,
                              void* d_out, int out_size, void* d_ws, size_t ws_size,
                              hipStream_t stream) {
  (void)in_sizes; (void)n_in; (void)out_size; (void)ws_size;
  const float* z  = (const float*)d_in[0];  // [2,1024,30]
  const float* Wr = (const float*)d_in[1];  // [60,64]
  const float* br = (const float*)d_in[2];  // [64]
  const float* Wa = (const float*)d_in[3];  // [64,30]
  const float* ba = (const float*)d_in[4];  // [30]
  float* out = (float*)d_out;               // [2,1024,30]

  float* c    = (float*)d_ws;               // [2048,64]
  float* s    = c + ROWS * HH;              // [2048,64]
  float* rsum = s + ROWS * HH;              // [2048,64]

  k_proj<<<dim3(ROWS / 16, HH / 16, 2), 32, 0, stream>>>(z, Wr, br, c, s);
  k_pair<<<dim3(ROWS / 16), 256, 0, stream>>>(c, s, rsum);
  k_out<<<dim3(ROWS / 16, 2), 32, 0, stream>>>(rsum, Wa, ba, out);
}